// Group_80582176408180
// MI455X (gfx1250) — compile-verified
//
#include <hip/hip_runtime.h>

#define NPTS   8192
#define NGROUP 512
#define GSIZE  32
#define NBATCH 16
#define BIGF   1e10f

typedef float v2f __attribute__((ext_vector_type(2)));
typedef float v8f __attribute__((ext_vector_type(8)));

// ---------------------------------------------------------------------------
// Kernel 1: farthest point sampling. One workgroup per batch, 1024 threads.
// Points live in LDS (96KB); running distances in registers (8 per thread).
// Writes selected center coordinates directly to the `centers` output region.
// ---------------------------------------------------------------------------
__global__ __launch_bounds__(1024) void fps_kernel(const float* __restrict__ xyz,
                                                   float* __restrict__ centers) {
  extern __shared__ float smem[];
  float* px = smem;                       // NPTS
  float* py = smem + NPTS;                // NPTS
  float* pz = smem + 2 * NPTS;            // NPTS
  float* wv = smem + 3 * NPTS;            // 32 per-wave max values
  int*   wi = (int*)(smem + 3 * NPTS + 32); // 32 per-wave argmax indices
  float* bs = smem + 3 * NPTS + 64;       // 3 broadcast coords of current pivot

  const int b    = blockIdx.x;
  const int tid  = threadIdx.x;
  const int lane = tid & 31;
  const int wave = tid >> 5;
  const float* src = xyz + (size_t)b * NPTS * 3;

  // Coalesced load of the cloud into LDS (SoA layout).
  for (int j = tid; j < NPTS * 3; j += 1024) {
    float v = src[j];
    int p = j / 3, c = j - 3 * p;
    (c == 0 ? px : (c == 1 ? py : pz))[p] = v;
  }
  __syncthreads();

  // Each thread owns 8 points at stride 1024 (bank-conflict-free).
  float x[8], y[8], z[8], dist[8];
#pragma unroll
  for (int i = 0; i < 8; ++i) {
    int p = i * 1024 + tid;
    x[i] = px[p]; y[i] = py[p]; z[i] = pz[p];
    dist[i] = BIGF;
  }

  if (tid == 0) {
    // Reference scan emits index 0 first.
    bs[0] = px[0]; bs[1] = py[0]; bs[2] = pz[0];
    float* co = centers + (size_t)b * NGROUP * 3;
    co[0] = px[0]; co[1] = py[0]; co[2] = pz[0];
  }
  __syncthreads();

  for (int s = 1; s < NGROUP; ++s) {
    float lx = bs[0], ly = bs[1], lz = bs[2];
    float mv = -1.0f; int mi = 0;
#pragma unroll
    for (int i = 0; i < 8; ++i) {
      float dx = x[i] - lx, dy = y[i] - ly, dzz = z[i] - lz;
      float d = dx * dx + dy * dy + dzz * dzz;
      d = fminf(dist[i], d);
      dist[i] = d;
      if (d > mv) { mv = d; mi = i * 1024 + tid; }  // strict > keeps lowest idx
    }
    // wave32 argmax, ties -> lowest index (matches jnp.argmax)
    for (int off = 16; off > 0; off >>= 1) {
      float ov = __shfl_xor(mv, off, 32);
      int   oi = __shfl_xor(mi, off, 32);
      if (ov > mv || (ov == mv && oi < mi)) { mv = ov; mi = oi; }
    }
    if (lane == 0) { wv[wave] = mv; wi[wave] = mi; }
    __syncthreads();
    if (tid < 32) {
      float v = wv[lane]; int i2 = wi[lane];
      for (int off = 16; off > 0; off >>= 1) {
        float ov = __shfl_xor(v,  off, 32);
        int   oi = __shfl_xor(i2, off, 32);
        if (ov > v || (ov == v && oi < i2)) { v = ov; i2 = oi; }
      }
      if (lane == 0) {
        float sx = px[i2], sy = py[i2], sz = pz[i2];
        bs[0] = sx; bs[1] = sy; bs[2] = sz;
        float* co = centers + ((size_t)b * NGROUP + s) * 3;
        co[0] = sx; co[1] = sy; co[2] = sz;
      }
    }
    __syncthreads();
  }
}

// ---------------------------------------------------------------------------
// Kernel 2: KNN (top-32 by squared distance) + gather + recenter.
// 256 threads = 8 waves; workgroup owns 8 centers; d2 (8 x 8192 f32 = 256KB)
// lives in LDS. The -2*c.p term is computed with V_WMMA_F32_16X16X4_F32:
// A = 16x4 center tile (rows 8..15 zero-padded), B = 4x16 point tile.
// A layout: lanes 0-15 hold {K0,K1} of row M=lane, lanes 16-31 hold {K2,K3}.
// B layout (C/D-analogous): v0 = {K0 | K2} across half-waves, v1 = {K1 | K3}.
// ---------------------------------------------------------------------------
__global__ __launch_bounds__(256) void knn_kernel(const float* __restrict__ xyz,
                                                  const float* __restrict__ centers,
                                                  float* __restrict__ nbh) {
  extern __shared__ float smem[];
  float* d2s  = smem;                 // 8 * NPTS
  float* ccs  = smem + 8 * NPTS;      // 16  (||c||^2, rows 8..15 = 0)
  float* cxyz = smem + 8 * NPTS + 16; // 48  (16 rows x 3 coords)

  const int b    = blockIdx.x >> 6;        // blockIdx.x / (NGROUP/8)
  const int g0   = (blockIdx.x & 63) * 8;
  const int tid  = threadIdx.x;
  const int lane = tid & 31;
  const int wave = tid >> 5;

  if (tid < 16) {
    float cx = 0.f, cy = 0.f, cz = 0.f;
    if (tid < 8) {
      const float* c = centers + ((size_t)b * NGROUP + g0 + tid) * 3;
      cx = c[0]; cy = c[1]; cz = c[2];
    }
    cxyz[tid * 3 + 0] = cx; cxyz[tid * 3 + 1] = cy; cxyz[tid * 3 + 2] = cz;
    ccs[tid] = cx * cx + cy * cy + cz * cz;
  }
  __syncthreads();

  // Build the A (center) fragment once; reused by all 64 WMMA iterations.
  const int mrow = lane & 15;
  v2f A;
  if (lane < 16) { A.x = cxyz[mrow * 3 + 0]; A.y = cxyz[mrow * 3 + 1]; }
  else           { A.x = cxyz[mrow * 3 + 2]; A.y = 0.0f; }

  float cc[8];
#pragma unroll
  for (int r = 0; r < 8; ++r) cc[r] = ccs[r];

  const float* pbase  = xyz + (size_t)b * NPTS * 3;
  const int    nstart = wave * (NPTS / 8);

  // Phase 1: fill d2s[center][point] for this wave's 1024-point span.
  for (int it = 0; it < (NPTS / 8) / 16; ++it) {
    int n = nstart + it * 16 + (lane & 15);
    const float* p = pbase + (size_t)n * 3;
    float pxv = p[0], pyv = p[1], pzv = p[2];
    float pp = pxv * pxv + pyv * pyv + pzv * pzv;
    v2f B;
    if (lane < 16) { B.x = pxv; B.y = pyv; }
    else           { B.x = pzv; B.y = 0.0f; }
    v8f acc = {};
    acc = __builtin_amdgcn_wmma_f32_16x16x4_f32(
        /*neg_a=*/false, A, /*neg_b=*/false, B,
        /*c_mod=*/(short)0, acc, /*reuse_a=*/false, /*reuse_b=*/false);
    // D layout: vgpr r, lanes 0-15 -> (M=r, N=lane); lanes 16-31 -> M=r+8 (pad).
    if (lane < 16) {
#pragma unroll
      for (int r = 0; r < 8; ++r)
        d2s[r * NPTS + n] = cc[r] + pp - 2.0f * acc[r];
    }
  }
  __syncthreads();

  // Phase 2: wave w selects 32 ascending minima for center g0+w.
  // Each lane caches the min of its 256-element chunk; per iteration only the
  // winning chunk is cooperatively rescanned (same-wave DS ops are in-order).
  volatile float* dz = d2s + wave * NPTS;
  float bv = BIGF; int bi = 0;
  {
    const float* dznc = d2s + wave * NPTS;
    const int base = lane * 256;
    for (int j = 0; j < 256; ++j) {
      float v = dznc[base + j];
      if (v < bv) { bv = v; bi = base + j; }   // strict < keeps lowest idx
    }
  }
  int my_sel = 0;   // lane k ends up holding the rank-k point index
  for (int k = 0; k < GSIZE; ++k) {
    float rv = bv; int ri = bi;
    for (int off = 16; off > 0; off >>= 1) {
      float ov = __shfl_xor(rv, off, 32);
      int   oi = __shfl_xor(ri, off, 32);
      if (ov < rv || (ov == rv && oi < ri)) { rv = ov; ri = oi; }
    }
    if (lane == k) my_sel = ri;
    if (lane == 0) dz[ri] = BIGF;              // mark selected
    const int c  = ri >> 8;                    // owning chunk
    const int rb = c * 256 + lane * 8;
    float nv = BIGF; int ni = rb;
#pragma unroll
    for (int j = 0; j < 8; ++j) {
      float v = dz[rb + j];
      if (v < nv) { nv = v; ni = rb + j; }
    }
    for (int off = 16; off > 0; off >>= 1) {
      float ov = __shfl_xor(nv, off, 32);
      int   oi = __shfl_xor(ni, off, 32);
      if (ov < nv || (ov == nv && oi < ni)) { nv = ov; ni = oi; }
    }
    if (lane == c) { bv = nv; bi = ni; }       // refresh invalidated cache
  }

  // Phase 3: gather + recenter + store. Lane k writes neighbor k.
  float cx = cxyz[wave * 3 + 0];
  float cy = cxyz[wave * 3 + 1];
  float cz = cxyz[wave * 3 + 2];
  const float* p = pbase + (size_t)my_sel * 3;
  float* o = nbh + (((size_t)b * NGROUP + g0 + wave) * GSIZE + lane) * 3;
  o[0] = p[0] - cx;
  o[1] = p[1] - cy;
  o[2] = p[2] - cz;
}

// ---------------------------------------------------------------------------
extern "C" void kernel_launch(void* const* d_in, const int* in_sizes, int n_in,
                              void* d_out, int out_size, void* d_ws, size_t ws_size,
                              hipStream_t stream) {
  (void)in_sizes; (void)n_in; (void)out_size; (void)d_ws; (void)ws_size;

  const float* xyz = (const float*)d_in[0];
  float* nbh       = (float*)d_out;                                   // (B,G,K,3)
  float* centers   = (float*)d_out + (size_t)NBATCH * NGROUP * GSIZE * 3; // (B,G,3)

  const size_t fps_smem = (size_t)(3 * NPTS + 64 + 4) * sizeof(float);      // ~96.3KB
  const size_t knn_smem = (size_t)(8 * NPTS + 16 + 48) * sizeof(float);     // ~256.3KB

  // Opt into >64KB dynamic LDS (immediate, idempotent; safe under capture).
  hipFuncSetAttribute((const void*)fps_kernel,
                      hipFuncAttributeMaxDynamicSharedMemorySize, (int)fps_smem);
  hipFuncSetAttribute((const void*)knn_kernel,
                      hipFuncAttributeMaxDynamicSharedMemorySize, (int)knn_smem);

  fps_kernel<<<NBATCH, 1024, fps_smem, stream>>>(xyz, centers);
  knn_kernel<<<NBATCH * (NGROUP / 8), 256, knn_smem, stream>>>(xyz, centers, nbh);
}